// RelationalGatedGraphConv_66202625900819
// MI455X (gfx1250) — compile-verified
//
#include <hip/hip_runtime.h>
#include <math.h>

typedef __attribute__((ext_vector_type(2))) float v2f;
typedef __attribute__((ext_vector_type(8))) float v8f;
typedef __attribute__((ext_vector_type(4))) int   v4i;

#define H 128
#define WPAD 132  // padded LDS row stride (floats): bank = (4n+k)%64 -> conflict free
#define TPW 4     // edge tiles (of 16 edges) per wave in msg kernel

#if defined(__has_builtin)
#if __has_builtin(__builtin_amdgcn_global_load_async_to_lds_b128)
#define HAVE_ASYNC_LDS 1
#endif
#if __has_builtin(__builtin_amdgcn_s_wait_asynccnt)
#define HAVE_WAIT_ASYNC_BUILTIN 1
#endif
#endif

// ---------------------------------------------------------------- zero agg
__global__ void zero_f32_kernel(float* __restrict__ p, long n) {
  long i = (long)blockIdx.x * blockDim.x + threadIdx.x;
  long stride = (long)gridDim.x * blockDim.x;
  for (; i < n; i += stride) p[i] = 0.0f;
}

// ---------------------------------------------------------------- messages + scatter
// grid: (ceil(Etiles/(8*TPW)), T), block: 256 (8 waves of 32).
// Block stages W_t (64KB) in LDS once (async-to-LDS when available), then each
// wave runs TPW tiles of 16 edges: C(16x128) = gathered src rows @ W_t^T via
// v_wmma_f32_16x16x4_f32, then bias + global_atomic_add_f32 scatter into agg[tgt].
__global__ __launch_bounds__(256) void msg_scatter_kernel(
    const float* __restrict__ hstate, const int* __restrict__ edges,
    const float* __restrict__ W, const float* __restrict__ b,
    float* __restrict__ agg, int E, int Etiles) {
  __shared__ float ldsW[H * WPAD];

  const int t   = blockIdx.y;
  const int tid = threadIdx.x;
  const float* Wt = W + (long)t * H * H;

#ifdef HAVE_ASYNC_LDS
  {
    typedef v4i __attribute__((address_space(1))) g_v4i;
    typedef v4i __attribute__((address_space(3))) l_v4i;
    g_v4i* gp = (g_v4i*)Wt;     // 32 v4i per global row
    l_v4i* lp = (l_v4i*)ldsW;   // 33 v4i per padded LDS row (WPAD/4)
    #pragma unroll
    for (int i = 0; i < (H * H) / (256 * 4); ++i) {   // 16 b128 transfers / thread
      const int flat4 = i * 256 + tid;                // float4 index
      const int row = flat4 >> 5, col4 = flat4 & 31;  // 32 float4 per row
      __builtin_amdgcn_global_load_async_to_lds_b128(
          gp + ((long)row * (H / 4) + col4),
          lp + (row * (WPAD / 4) + col4), 0, 0);
    }
#ifdef HAVE_WAIT_ASYNC_BUILTIN
    __builtin_amdgcn_s_wait_asynccnt(0);
#else
    asm volatile("s_wait_asynccnt 0" ::: "memory");
#endif
  }
#else
  #pragma unroll 8
  for (int i = 0; i < (H * H) / 256; ++i) {
    const int idx = i * 256 + tid;
    ldsW[(idx >> 7) * WPAD + (idx & (H - 1))] = Wt[idx];
  }
#endif

  const int wave = tid >> 5;
  const int lane = tid & 31;
  const int m    = lane & 15;   // M row / N col within 16-tile
  const int half = lane >> 4;   // k sub-offset selector

  __syncthreads();

  // per-lane output-column bias (depends only on nn and m)
  const float* bt = b + (long)t * H;
  float bias[8];
  #pragma unroll
  for (int nn = 0; nn < 8; ++nn) bias[nn] = bt[nn * 16 + m];

  const int tileBase = (blockIdx.x * 8 + wave) * TPW;
  #pragma unroll 1
  for (int it = 0; it < TPW; ++it) {
    const int tile = tileBase + it;
    if (tile >= Etiles) break;  // wave-uniform
    const int e = tile * 16 + m;
    int srcIdx = 0, tgtIdx = -1;
    if (e < E) {
      srcIdx = edges[(long)t * 2 * E + e];       // edge_lists[t][0][e]
      tgtIdx = edges[(long)t * 2 * E + E + e];   // edge_lists[t][1][e]
    }
    const float* arow = hstate + (long)srcIdx * H;

    v8f acc[8] = {};
    #pragma unroll 4
    for (int k0 = 0; k0 < H; k0 += 4) {
      const int k = k0 + half * 2;
      v2f a = *(const v2f*)(arow + k);             // gathered A fragment (8B aligned)
      #pragma unroll
      for (int nn = 0; nn < 8; ++nn) {
        v2f bf = *(const v2f*)&ldsW[(nn * 16 + m) * WPAD + k];
        acc[nn] = __builtin_amdgcn_wmma_f32_16x16x4_f32(
            false, a, false, bf, (short)0, acc[nn], false, false);
      }
    }

    // scatter: C vgpr j holds (M = j or j+8, N = nn*16 + m); tgt via shuffle
    #pragma unroll
    for (int j = 0; j < 8; ++j) {
      const int mo = half ? (j + 8) : j;
      const int tg = __shfl(tgtIdx, mo, 32);       // lane mo holds edge mo's target
      if (tg >= 0) {
        float* dst = agg + (long)tg * H + m;
        #pragma unroll
        for (int nn = 0; nn < 8; ++nn)
          atomicAdd(dst + nn * 16, acc[nn][j] + bias[nn]);
      }
    }
  }
}

// ---------------------------------------------------------------- GRU update
// grid: ceil(Ntiles/8), block 256 (8 waves). Each wave: 16 nodes.
// gi = agg @ w_ih^T + b_ih ; gh = h @ w_hh^T + b_hh ; gated update -> out.
__global__ __launch_bounds__(256) void gru_update_kernel(
    const float* __restrict__ agg, const float* __restrict__ hstate,
    const float* __restrict__ w_ih, const float* __restrict__ w_hh,
    const float* __restrict__ b_ih, const float* __restrict__ b_hh,
    float* __restrict__ out, int N, int Ntiles) {
  __shared__ float ldsA[8][16 * WPAD];
  __shared__ float ldsH[8][16 * WPAD];

  const int tid  = threadIdx.x;
  const int wave = tid >> 5;
  const int lane = tid & 31;
  const int m    = lane & 15;
  const int half = lane >> 4;
  const int tile = blockIdx.x * 8 + wave;

  if (tile < Ntiles) {
    const int node0 = tile * 16;
    #pragma unroll 4
    for (int r = 0; r < 16; ++r) {
      int row = node0 + r;
      if (row >= N) row = N - 1;  // clamp: staged but never stored
      float4 av = ((const float4*)(agg    + (long)row * H))[lane];
      float4 hv = ((const float4*)(hstate + (long)row * H))[lane];
      *(float4*)&ldsA[wave][r * WPAD + lane * 4] = av;  // (528r+16l) bytes: 16B aligned
      *(float4*)&ldsH[wave][r * WPAD + lane * 4] = hv;
    }
  }
  __syncthreads();
  if (tile >= Ntiles) return;  // wave-uniform exit
  const int node0 = tile * 16;

  #pragma unroll 1
  for (int j = 0; j < 8; ++j) {  // 16-column tile of the 128 output cols
    v8f ir = {}, iz = {}, in_ = {}, hr = {}, hz = {}, hn = {};
    const int nr = j * 16 + m;   // weight row for this lane's output column
    const float* wir = w_ih + (long)nr * H;
    const float* wiz = w_ih + (long)(nr + H) * H;
    const float* win = w_ih + (long)(nr + 2 * H) * H;
    const float* whr = w_hh + (long)nr * H;
    const float* whz = w_hh + (long)(nr + H) * H;
    const float* whn = w_hh + (long)(nr + 2 * H) * H;

    #pragma unroll 4
    for (int k0 = 0; k0 < H; k0 += 4) {
      const int k = k0 + half * 2;
      v2f aA = *(const v2f*)&ldsA[wave][m * WPAD + k];
      v2f aH = *(const v2f*)&ldsH[wave][m * WPAD + k];
      v2f br = *(const v2f*)(wir + k);
      v2f bz = *(const v2f*)(wiz + k);
      v2f bn = *(const v2f*)(win + k);
      v2f cr = *(const v2f*)(whr + k);
      v2f cz = *(const v2f*)(whz + k);
      v2f cn = *(const v2f*)(whn + k);
      ir  = __builtin_amdgcn_wmma_f32_16x16x4_f32(false, aA, false, br, (short)0, ir,  false, false);
      iz  = __builtin_amdgcn_wmma_f32_16x16x4_f32(false, aA, false, bz, (short)0, iz,  false, false);
      in_ = __builtin_amdgcn_wmma_f32_16x16x4_f32(false, aA, false, bn, (short)0, in_, false, false);
      hr  = __builtin_amdgcn_wmma_f32_16x16x4_f32(false, aH, false, cr, (short)0, hr,  false, false);
      hz  = __builtin_amdgcn_wmma_f32_16x16x4_f32(false, aH, false, cz, (short)0, hz,  false, false);
      hn  = __builtin_amdgcn_wmma_f32_16x16x4_f32(false, aH, false, cn, (short)0, hn,  false, false);
    }

    const int col = j * 16 + m;
    const float bir = b_ih[col], biz = b_ih[col + H], bin = b_ih[col + 2 * H];
    const float bhr = b_hh[col], bhz = b_hh[col + H], bhn = b_hh[col + 2 * H];

    #pragma unroll
    for (int e = 0; e < 8; ++e) {
      const int mo  = half ? (e + 8) : e;
      const int row = node0 + mo;
      const float r  = 1.0f / (1.0f + __expf(-(ir[e] + bir + hr[e] + bhr)));
      const float z  = 1.0f / (1.0f + __expf(-(iz[e] + biz + hz[e] + bhz)));
      const float hv = ldsH[wave][mo * WPAD + col];
      const float ng = tanhf(in_[e] + bin + r * (hn[e] + bhn));
      const float o  = (1.0f - z) * ng + z * hv;
      if (row < N) out[(long)row * H + col] = o;
    }
  }
}

// ---------------------------------------------------------------- launcher
extern "C" void kernel_launch(void* const* d_in, const int* in_sizes, int n_in,
                              void* d_out, int out_size, void* d_ws, size_t ws_size,
                              hipStream_t stream) {
  const float* node_states = (const float*)d_in[0];
  const int*   edges       = (const int*)d_in[1];
  const float* W           = (const float*)d_in[2];
  const float* b           = (const float*)d_in[3];
  const float* w_ih        = (const float*)d_in[4];
  const float* w_hh        = (const float*)d_in[5];
  const float* b_ih        = (const float*)d_in[6];
  const float* b_hh        = (const float*)d_in[7];

  const int N = in_sizes[0] / H;           // 100000
  const int T = in_sizes[2] / (H * H);     // 4
  const int E = in_sizes[1] / (2 * T);     // 400000

  float* agg = (float*)d_ws;               // N*H f32 = 51.2 MB scratch
  const long aggN = (long)N * H;
  zero_f32_kernel<<<2048, 256, 0, stream>>>(agg, aggN);

  const int Etiles = (E + 15) / 16;
  dim3 gridB((Etiles + 8 * TPW - 1) / (8 * TPW), T);
  msg_scatter_kernel<<<gridB, 256, 0, stream>>>(node_states, edges, W, b, agg, E, Etiles);

  const int Ntiles = (N + 15) / 16;
  gru_update_kernel<<<(Ntiles + 7) / 8, 256, 0, stream>>>(
      agg, node_states, w_ih, w_hh, b_ih, b_hh, (float*)d_out, N, Ntiles);
}